// GridSample_88459146428541
// MI455X (gfx1250) — compile-verified
//
#include <hip/hip_runtime.h>

// grid_sample bilinear, zeros padding.
// features: (8, 256, 256, 64) f32   grid: (8, 512, 512, 2) f32 in [-1,1]
// out:      (8, 512, 512, 64) f32
//
// Memory-bound gather: keep 134MB features resident in the 192MB L2 by
// streaming the 537MB output with non-temporal stores and the grid with
// non-temporal loads. WMMA is inapplicable (no shared operand across any
// 16x16 tile -- every pixel gathers its own 4 feature rows).

typedef float v4f __attribute__((ext_vector_type(4)));
typedef float v2f __attribute__((ext_vector_type(2)));

namespace {
constexpr int kBlock = 256;
constexpr int kHF = 256;       // feature H
constexpr int kWF = 256;       // feature W
constexpr int kC4 = 16;        // C/4 = 64/4 float4 groups per pixel
// output pixels: 8 * 512 * 512 = 1 << 21
}

__global__ __launch_bounds__(kBlock) void GridSampleKernel(
    const float* __restrict__ features,
    const v2f*  __restrict__ grid2,
    v4f*        __restrict__ out4)
{
    const int gid = blockIdx.x * kBlock + (int)threadIdx.x;
    const int pix = gid >> 4;          // global output pixel index
    const int c4  = gid & 15;          // float4 channel group within pixel
    const int b   = pix >> 18;         // 512*512 = 2^18 pixels per batch

    // Grid is streamed exactly once: NT load, and prefetch the stream ahead
    // (gfx1250 global_prefetch_b8; speculative, OOB-safe per ISA).
    v2f g = __builtin_nontemporal_load(&grid2[pix]);
    __builtin_prefetch((const void*)(&grid2[pix + 8192]), 0, 1);

    // Reference reverses the last grid axis: grid[...,0] -> x (W axis),
    // grid[...,1] -> y (H axis);  coord = (g + 1) * dim * 0.5 - 0.5.
    const float x = (g.x + 1.0f) * 128.0f - 0.5f;
    const float y = (g.y + 1.0f) * 128.0f - 0.5f;
    const float fx = floorf(x), fy = floorf(y);
    const int x0 = (int)fx, y0 = (int)fy;
    const int x1 = x0 + 1,  y1 = y0 + 1;
    const float tx = x - fx, ty = y - fy;

    float w00 = (1.0f - ty) * (1.0f - tx);   // (y0, x0)
    float w01 = (1.0f - ty) * tx;            // (y0, x1)
    float w10 = ty * (1.0f - tx);            // (y1, x0)
    float w11 = ty * tx;                     // (y1, x1)

    // Zeros padding == zero the weight when the neighbor is out of bounds,
    // and clamp the actually-loaded coordinate (branchless, no divergence).
    const bool vx0 = (unsigned)x0 < (unsigned)kWF;
    const bool vx1 = (unsigned)x1 < (unsigned)kWF;
    const bool vy0 = (unsigned)y0 < (unsigned)kHF;
    const bool vy1 = (unsigned)y1 < (unsigned)kHF;
    w00 = (vy0 && vx0) ? w00 : 0.0f;
    w01 = (vy0 && vx1) ? w01 : 0.0f;
    w10 = (vy1 && vx0) ? w10 : 0.0f;
    w11 = (vy1 && vx1) ? w11 : 0.0f;

    const int cx0 = min(max(x0, 0), kWF - 1);
    const int cx1 = min(max(x1, 0), kWF - 1);
    const int cy0 = min(max(y0, 0), kHF - 1);
    const int cy1 = min(max(y1, 0), kHF - 1);

    // Feature addressing in float4 units: b*(256*256*16) + (y*256+x)*16 + c4
    const v4f* __restrict__ f4 = (const v4f*)features;
    const int fb = (b << 20) + c4;
    const int r00 = fb + (((cy0 << 8) + cx0) << 4);
    const int r01 = fb + (((cy0 << 8) + cx1) << 4);
    const int r10 = fb + (((cy1 << 8) + cx0) << 4);
    const int r11 = fb + (((cy1 << 8) + cx1) << 4);

    // Regular (RT) loads: features stay hot in the 192MB L2.
    const v4f v00 = f4[r00];
    const v4f v01 = f4[r01];
    const v4f v10 = f4[r10];
    const v4f v11 = f4[r11];

    v4f acc = v00 * w00;
    acc += v01 * w01;
    acc += v10 * w10;
    acc += v11 * w11;

    // Output is written once and never re-read: NT store so the 537MB stream
    // does not evict the L2-resident feature map.
    __builtin_nontemporal_store(acc, &out4[(pix << 4) + c4]);
}

extern "C" void kernel_launch(void* const* d_in, const int* in_sizes, int n_in,
                              void* d_out, int out_size, void* d_ws, size_t ws_size,
                              hipStream_t stream) {
    (void)in_sizes; (void)n_in; (void)out_size; (void)d_ws; (void)ws_size;
    const float* features = (const float*)d_in[0];
    const v2f*   grid2    = (const v2f*)d_in[1];
    v4f*         out4     = (v4f*)d_out;

    const long long pixels  = 8LL * 512 * 512;        // 2,097,152
    const long long threads = pixels * 16;            // one thread per float4
    const int blocks = (int)(threads / kBlock);       // 131,072

    GridSampleKernel<<<blocks, kBlock, 0, stream>>>(features, grid2, out4);
}